// Canny_60129542552
// MI455X (gfx1250) — compile-verified
//
#include <hip/hip_runtime.h>
#include <hip/hip_bf16.h>

namespace {
constexpr int kB = 16, kC = 3, kH = 512, kW = 512;
constexpr int kHH = kB * kH;            // 8192-row strip
constexpr int kNPix = kHH * kW;         // 4194304
constexpr int kHyst = 48;               // chained hysteresis launches
constexpr float kTG22 = 0.41421356237309515f;  // rounds to f32 like jnp cast
}

// ---- CDNA5 async global->LDS DMA helpers (gfx1250, ASYNCcnt path) ----------
__device__ __forceinline__ unsigned lds_off(const void* p) {
  // ISA 10.2: flat LDS address low 32 bits are the LDS offset.
  return (unsigned)(unsigned long long)p;
}
__device__ __forceinline__ void async_ld_b32(const void* gp, const void* lp) {
  asm volatile("global_load_async_to_lds_b32 %0, %1, off"
               :: "v"(lds_off(lp)), "v"(gp) : "memory");
}
__device__ __forceinline__ void async_wait0() {
  asm volatile("s_wait_asynccnt 0" ::: "memory");
}

// ---- Pass 1: quantize + 3x3 Sobel + channel argmax + direction code --------
__global__ __launch_bounds__(256)
void canny_sobel(const float* __restrict__ xin,
                 float* __restrict__ mag, unsigned char* __restrict__ dir) {
  __shared__ __align__(16) float T[3][10][34];   // 3 channels, 8x32 tile + halo
  const int tx = threadIdx.x, ty = threadIdx.y;
  const int tid = ty * 32 + tx;
  const int col0 = blockIdx.x * 32;
  const int row0 = blockIdx.y * 8;

  // Stage halo tile via async DMA (replicate border like cv2 / jnp edge-pad).
  for (int e = tid; e < 3 * 10 * 34; e += 256) {
    int c = e / 340;
    int rem = e - c * 340;
    int rr = rem / 34;
    int cc = rem - rr * 34;
    int r = row0 - 1 + rr;  r = r < 0 ? 0 : (r > kHH - 1 ? kHH - 1 : r);
    int cl = col0 - 1 + cc; cl = cl < 0 ? 0 : (cl > kW - 1 ? kW - 1 : cl);
    const float* gp = xin + ((size_t)((r >> 9) * 3 + c) << 18)
                          + ((size_t)(r & 511) << 9) + cl;
    async_ld_b32(gp, &T[c][rr][cc]);
  }
  async_wait0();
  __syncthreads();

  // Quantize in place: clip(floor((x+1)*0.5*255), 0, 255); values stay exact ints.
  for (int e = tid; e < 3 * 10 * 34; e += 256) {
    int c = e / 340; int rem = e - c * 340; int rr = rem / 34; int cc = rem - rr * 34;
    float q = floorf((T[c][rr][cc] + 1.0f) * 0.5f * 255.0f);
    T[c][rr][cc] = fminf(fmaxf(q, 0.0f), 255.0f);
  }
  __syncthreads();

  const int ly = ty + 1, lx = tx + 1;
  float bm = -1.0f, bgx = 0.0f, bgy = 0.0f;
#pragma unroll
  for (int c = 0; c < 3; ++c) {
    float a00 = T[c][ly-1][lx-1], a01 = T[c][ly-1][lx], a02 = T[c][ly-1][lx+1];
    float a10 = T[c][ly  ][lx-1],                       a12 = T[c][ly  ][lx+1];
    float a20 = T[c][ly+1][lx-1], a21 = T[c][ly+1][lx], a22 = T[c][ly+1][lx+1];
    float gx = (a02 + 2.0f * a12 + a22) - (a00 + 2.0f * a10 + a20);
    float gy = (a20 + 2.0f * a21 + a22) - (a00 + 2.0f * a01 + a02);
    float m = fabsf(gx) + fabsf(gy);          // exact (integer-valued, small)
    if (m > bm) { bm = m; bgx = gx; bgy = gy; }  // strict > == first argmax
  }
  float ax = fabsf(bgx), ay = fabsf(bgy);
  unsigned char d;
  if (ay < kTG22 * ax)      d = 0;                         // horizontal
  else if (ay * kTG22 > ax) d = 1;                         // vertical
  else d = (bgx * bgy >= 0.0f) ? (unsigned char)2 : (unsigned char)3;  // diag
  const int gi = (row0 + ty) * kW + (col0 + tx);
  mag[gi] = bm;
  dir[gi] = d;
}

// ---- Pass 2: NMS (zero-padded neighbors) + double threshold ----------------
__global__ __launch_bounds__(256)
void canny_nms(const float* __restrict__ mag, const unsigned char* __restrict__ dir,
               unsigned char* __restrict__ strong, unsigned char* __restrict__ weak) {
  int i = blockIdx.x * 256 + threadIdx.x;
  if (i >= kNPix) return;
  int r = i >> 9, c = i & 511;
  float m = mag[i];
  auto M = [&](int rr, int cc) -> float {
    return ((unsigned)rr < (unsigned)kHH && (unsigned)cc < (unsigned)kW)
               ? mag[rr * kW + cc] : 0.0f;
  };
  float n1, n2;
  switch (dir[i]) {
    case 0:  n1 = M(r, c - 1);     n2 = M(r, c + 1);     break;
    case 1:  n1 = M(r - 1, c);     n2 = M(r + 1, c);     break;
    case 2:  n1 = M(r - 1, c - 1); n2 = M(r + 1, c + 1); break;
    default: n1 = M(r - 1, c + 1); n2 = M(r + 1, c - 1); break;
  }
  bool keep = (m > n1) && (m >= n2);
  strong[i] = (keep && m > 200.0f) ? 1 : 0;
  weak[i]   = (keep && m > 100.0f) ? 1 : 0;
}

__global__ void canny_init_flags(int* __restrict__ flags) {
  int i = threadIdx.x;
  if (i < 64) flags[i] = (i == 0) ? 1 : 0;
}

// ---- Hysteresis: tiled in-LDS fixed point, flag-chained across launches ----
__global__ __launch_bounds__(256)
void canny_hyst(unsigned char* __restrict__ strong,
                const unsigned char* __restrict__ weak,
                int* __restrict__ flags, int iter) {
  if (flags[iter - 1] == 0) return;   // converged earlier: launch is a no-op
  __shared__ __align__(16) unsigned char SB[34][40];  // strong + halo (col+4 base)
  __shared__ __align__(16) unsigned char WK[32][32];  // weak interior
  __shared__ int s_changed;
  const int tx = threadIdx.x, ty = threadIdx.y;
  const int tid = ty * 32 + tx;
  const int col0 = blockIdx.x * 32, row0 = blockIdx.y * 32;

  // Interior: one aligned u32 async DMA per thread for weak and strong.
  {
    int rowi = tid >> 3;        // 0..31
    int c4 = (tid & 7) * 4;     // 0,4,...,28
    async_ld_b32(weak   + (size_t)(row0 + rowi) * kW + col0 + c4, &WK[rowi][c4]);
    async_ld_b32(strong + (size_t)(row0 + rowi) * kW + col0 + c4, &SB[1 + rowi][4 + c4]);
  }
  // Halo ring: zero outside the strip (reference dilates with zero padding).
  if (tid < 132) {
    int gy, gx; unsigned char* dst;
    if (tid < 34)       { gy = row0 - 1 + tid;         gx = col0 - 1;            dst = &SB[tid][3]; }
    else if (tid < 68)  { gy = row0 - 1 + (tid - 34);  gx = col0 + 32;           dst = &SB[tid - 34][36]; }
    else if (tid < 100) { gy = row0 - 1;               gx = col0 + (tid - 68);   dst = &SB[0][4 + (tid - 68)]; }
    else                { gy = row0 + 32;              gx = col0 + (tid - 100);  dst = &SB[33][4 + (tid - 100)]; }
    unsigned char v = 0;
    if ((unsigned)gy < (unsigned)kHH && (unsigned)gx < (unsigned)kW)
      v = strong[(size_t)gy * kW + gx];
    *dst = v;
  }
  async_wait0();
  __syncthreads();

  // In-tile Gauss-Seidel fixed point (monotone 0->1; byte races are benign).
  for (int it = 0; it < 40; ++it) {
    if (tid == 0) s_changed = 0;
    __syncthreads();
    bool me = false;
#pragma unroll
    for (int k = 0; k < 4; ++k) {
      int y = ty + 8 * k;
      if (!SB[1 + y][4 + tx] && WK[y][tx]) {
        int nb = SB[y][3 + tx] | SB[y][4 + tx] | SB[y][5 + tx]
               | SB[1 + y][3 + tx] | SB[1 + y][5 + tx]
               | SB[2 + y][3 + tx] | SB[2 + y][4 + tx] | SB[2 + y][5 + tx];
        if (nb) { SB[1 + y][4 + tx] = 1; me = true; }
      }
    }
    if (me) s_changed = 1;
    __syncthreads();
    int ch = s_changed;
    __syncthreads();
    if (!ch) break;
  }

  // Write back owned pixels; chain convergence flag for the next launch.
  bool diff = false;
#pragma unroll
  for (int k = 0; k < 4; ++k) {
    int y = ty + 8 * k;
    size_t gi = (size_t)(row0 + y) * kW + col0 + tx;
    unsigned char nv = SB[1 + y][4 + tx];
    if (nv != strong[gi]) { strong[gi] = nv; diff = true; }
  }
  if (diff) flags[iter] = 1;
}

// ---- Output: {0,1} -> {-1,+1}, broadcast over 3 channel planes -------------
__global__ __launch_bounds__(256)
void canny_out(const unsigned char* __restrict__ strong, float* __restrict__ out) {
  int i = blockIdx.x * 256 + threadIdx.x;
  if (i >= kNPix) return;
  float v = strong[i] ? 1.0f : -1.0f;
  int b = i >> 18;                       // strip row = b*512 + y
  int yx = i & (kH * kW - 1);
  size_t base = (size_t)b * 3 * kH * kW + yx;
  out[base] = v;
  out[base + kH * kW] = v;
  out[base + 2 * kH * kW] = v;
}

extern "C" void kernel_launch(void* const* d_in, const int* in_sizes, int n_in,
                              void* d_out, int out_size, void* d_ws, size_t ws_size,
                              hipStream_t stream) {
  (void)in_sizes; (void)n_in; (void)out_size; (void)ws_size;
  const float* xin = (const float*)d_in[0];
  float* out = (float*)d_out;
  unsigned char* ws = (unsigned char*)d_ws;

  unsigned char* strong = ws;
  unsigned char* weak   = ws + (size_t)kNPix;
  unsigned char* dir    = ws + 2 * (size_t)kNPix;
  float* mag            = (float*)(ws + 3 * (size_t)kNPix);
  int* flags            = (int*)(ws + 3 * (size_t)kNPix + (size_t)kNPix * 4);

  dim3 blk(32, 8);
  canny_sobel<<<dim3(kW / 32, kHH / 8), blk, 0, stream>>>(xin, mag, dir);
  canny_nms<<<dim3(kNPix / 256), dim3(256), 0, stream>>>(mag, dir, strong, weak);
  canny_init_flags<<<1, 64, 0, stream>>>(flags);
  for (int it = 1; it <= kHyst; ++it)
    canny_hyst<<<dim3(kW / 32, kHH / 32), blk, 0, stream>>>(strong, weak, flags, it);
  canny_out<<<dim3(kNPix / 256), dim3(256), 0, stream>>>(strong, out);
}